// MNIST_QNN_Classifier_730144440592
// MI455X (gfx1250) — compile-verified
//
#include <hip/hip_runtime.h>

// ============================================================================
// QNN expectation value on gfx1250 (MI455X, wave32, WMMA).
//
// expv = sum_b w[b] * ( v0^2 + v1^2 - v2^2 - v3^2 ),  v = column 0 of
// Q_b = M3 M2 M1 M0, M_l = CNOT * (RY(th_l0) (x) RY(th_l1))  (real 4x4),
// w[b] = prod over 12 tree levels of cos^2/sin^2(alpha/2).
// Output: 64 copies of expv.
//
// WMMA mapping: 4 blocks packed as a block-diagonal 16x16; U <- M_l * U done
// as 4 chained V_WMMA_F32_16X16X4_F32 (chunk k: A = M[:,4k:4k+4] (rows of
// block k only), B = U[4k:4k+4,:]), which reproduces the block-diag product
// exactly in fp32 (reference precision).
// ============================================================================

typedef float v2f __attribute__((ext_vector_type(2)));
typedef float v8f __attribute__((ext_vector_type(8)));

#define NB             4096          // qnn blocks (= 2^12)
#define NGROUPS        (NB / 4)      // 1024 groups of 4 blocks
#define WGS            256
#define NWG            16
#define WAVES_PER_WG   (WGS / 32)
#define TOTAL_WAVES    (NWG * WAVES_PER_WG)        // 128
#define GROUPS_PER_WAVE (NGROUPS / TOTAL_WAVES)    // 8

#define USTRIDE 17                   // padded row stride for U scratch (bank-conflict free)
#define LDS_PER_WAVE (64 + 256 + 16 * USTRIDE)     // CS + M + U = 592 floats

__global__ __launch_bounds__(WGS)
void qnn_main(const float* __restrict__ alpha,
              const float* __restrict__ thetas,
              float* __restrict__ partials) {
  __shared__ float lds[WAVES_PER_WG * LDS_PER_WAVE];
  __shared__ float red[WGS];

  const int tid   = threadIdx.x;
  const int lane  = tid & 31;
  const int wave  = tid >> 5;
  const int gwave = blockIdx.x * WAVES_PER_WG + wave;

  float* CS   = lds + wave * LDS_PER_WAVE;   // 64: (c,s) per theta, idx = 2*lane
  float* Mlds = CS + 64;                     // 256: [layer][blk][row][col]
  float* Ulds = CS + 320;                    // 16 x USTRIDE: U scratch

  const int n     = lane & 15;               // col idx (B/C frags), row idx (A frag)
  const int half  = lane >> 4;               // K-half selector
  const int mbase = half ? 8 : 0;            // C-frag row base

  float acc = 0.0f;

  for (int it = 0; it < GROUPS_PER_WAVE; ++it) {
    const int g  = gwave + TOTAL_WAVES * it; // group id in [0, 1024)
    const int b0 = g * 4;                    // first qnn block of this group

    // Prefetch next group's thetas (global_prefetch_b8).
    if (it + 1 < GROUPS_PER_WAVE)
      __builtin_prefetch(thetas + (b0 + TOTAL_WAVES * 4) * 8 + lane, 0, 3);

    // -------- phase 1: one theta per lane -> (cos, sin) of t/2 ------------
    // lane = beta*8 + l*2 + q  (beta = block-in-group, l = layer, q = qubit)
    {
      float t = thetas[b0 * 8 + lane];
      float s, c;
      __sincosf(0.5f * t, &s, &c);
      CS[2 * lane + 0] = c;
      CS[2 * lane + 1] = s;
    }
    __syncthreads();

    // -------- phase 2: build M_l = CNOT * (R0 (x) R1), 4 layers x 4 blocks
    for (int e = lane; e < 256; e += 32) {
      int l    = e >> 6;
      int rm   = e & 63;
      int beta = rm >> 4;
      int i    = (rm >> 2) & 3;
      int j    = rm & 3;
      int p    = i ^ (i >> 1);               // CNOT row permutation (2<->3)
      int cb   = 2 * (beta * 8 + l * 2);
      float c0 = CS[cb + 0], s0 = CS[cb + 1];
      float c1 = CS[cb + 2], s1 = CS[cb + 3];
      int a0 = p >> 1, b0c = j >> 1;
      int a1 = p & 1,  b1c = j & 1;
      // RY = [[c,-s],[s,c]]
      float r0 = (a0 == b0c) ? c0 : (a0 ? s0 : -s0);
      float r1 = (a1 == b1c) ? c1 : (a1 ? s1 : -s1);
      Mlds[e] = r0 * r1;
    }
    __syncthreads();

    // -------- phase 3: seed C fragment with blockdiag(M_0) ----------------
    // C layout: elem (m,n): lane = n + (m>=8 ? 16 : 0), vgpr = m&7
    v8f C;
#pragma unroll
    for (int j = 0; j < 8; ++j) {
      int m = mbase + j;
      C[j] = ((m >> 2) == (n >> 2))
                 ? Mlds[(n >> 2) * 16 + (m & 3) * 4 + (n & 3)]
                 : 0.0f;
    }

    // -------- phase 4: U <- M_l * U via v_wmma_f32_16x16x4_f32 ------------
    for (int l = 1; l < 4; ++l) {
      // dump current U (C layout) to LDS (row-major, padded)
#pragma unroll
      for (int j = 0; j < 8; ++j) Ulds[(mbase + j) * USTRIDE + n] = C[j];
      __syncthreads();

      v8f Cn = {0.f, 0.f, 0.f, 0.f, 0.f, 0.f, 0.f, 0.f};
#pragma unroll
      for (int k = 0; k < 4; ++k) {
        v2f a, b;
#pragma unroll
        for (int jj = 0; jj < 2; ++jj) {
          // A frag: elem (m,kk): lane = m + (kk>=2?16:0), vgpr = kk&1
          // chunk k columns live only in rows of block k (block-diagonal)
          a[jj] = ((n >> 2) == k)
                      ? Mlds[l * 64 + k * 16 + (n & 3) * 4 + (2 * half + jj)]
                      : 0.0f;
          // B frag: elem (kk,n): lane = n + (kk>=2?16:0), vgpr = kk&1
          b[jj] = Ulds[(4 * k + 2 * half + jj) * USTRIDE + n];
        }
        Cn = __builtin_amdgcn_wmma_f32_16x16x4_f32(
            /*neg_a=*/false, a, /*neg_b=*/false, b,
            /*c_mod=*/(short)0, Cn, /*reuse_a=*/false, /*reuse_b=*/false);
      }
      C = Cn;
      __syncthreads();
    }

    // -------- phase 5: extract col 0 of each block, weight, accumulate ----
#pragma unroll
    for (int j = 0; j < 8; ++j) Ulds[(mbase + j) * USTRIDE + n] = C[j];
    __syncthreads();

    if (lane < 4) {
      const int beta = lane;
      const int col  = 4 * beta;
      float v0 = Ulds[(col + 0) * USTRIDE + col];
      float v1 = Ulds[(col + 1) * USTRIDE + col];
      float v2 = Ulds[(col + 2) * USTRIDE + col];
      float v3 = Ulds[(col + 3) * USTRIDE + col];
      float q  = v0 * v0 + v1 * v1 - v2 * v2 - v3 * v3;

      // tree weight w[b] = prod_l cos^2/sin^2(alpha_node / 2)
      const int b = b0 + beta;
      float w = 1.0f;
#pragma unroll
      for (int l = 0; l < 12; ++l) {
        int node  = b >> (12 - l);
        float ang = alpha[(1 << l) - 1 + node];
        float sh, ch;
        __sincosf(0.5f * ang, &sh, &ch);
        float f = ((b >> (11 - l)) & 1) ? sh : ch;
        w *= f * f;
      }
      acc += w * q;
    }
    __syncthreads();
  }

  // -------- deterministic block reduction --------------------------------
  red[tid] = acc;
  __syncthreads();
  if (tid == 0) {
    float s = 0.0f;
    for (int i = 0; i < WGS; ++i) s += red[i];
    partials[blockIdx.x] = s;
  }
}

// Sum the 16 per-workgroup partials in fixed order; broadcast to all 64 outputs.
__global__ void qnn_finalize(const float* __restrict__ partials,
                             float* __restrict__ out) {
  float s = 0.0f;
#pragma unroll
  for (int i = 0; i < NWG; ++i) s += partials[i];
  out[threadIdx.x] = s;
}

extern "C" void kernel_launch(void* const* d_in, const int* in_sizes, int n_in,
                              void* d_out, int out_size, void* d_ws, size_t ws_size,
                              hipStream_t stream) {
  // d_in[0] = x (64,4) f32  -- unused: circuit is independent of x
  // d_in[1] = alpha (4095,) f32
  // d_in[2] = qnn_thetas (4096, 8) f32
  const float* alpha  = (const float*)d_in[1];
  const float* thetas = (const float*)d_in[2];
  float* partials = (float*)d_ws;   // 16 floats of scratch

  qnn_main<<<NWG, WGS, 0, stream>>>(alpha, thetas, partials);
  qnn_finalize<<<1, 64, 0, stream>>>(partials, (float*)d_out);
}